// RoutingLoRASoftMoe_77309411328461
// MI455X (gfx1250) — compile-verified
//
#include <hip/hip_runtime.h>
#include <math.h>

#define BS    4
#define SEQ   1024
#define DIN   1024
#define DOUT  1024
#define NSK   8
#define RANK  4
#define NKR   (NSK*RANK)      // 32
#define NTOK  (BS*SEQ)        // 4096

typedef __attribute__((ext_vector_type(2))) float v2f;
typedef __attribute__((ext_vector_type(8))) float v8f;
typedef unsigned int v4u __attribute__((ext_vector_type(4)));
typedef int v4i __attribute__((ext_vector_type(4)));
typedef int v8i __attribute__((ext_vector_type(8)));

// ---------------------------------------------------------------------------
// Kernel 1: per-batch mask params.  pad_len = first pad==1 (left padding),
// thresh = first_inst + n_inst  (end of instruction prefix).
// ---------------------------------------------------------------------------
__global__ void k_params(const int* __restrict__ inst, const int* __restrict__ pad,
                         int* __restrict__ params) {
    int b = blockIdx.x;
    int t = threadIdx.x;
    __shared__ int s_fi, s_fp, s_ni;
    if (t == 0) { s_fi = SEQ; s_fp = SEQ; s_ni = 0; }
    __syncthreads();
    int fi = SEQ, fp = SEQ, ni = 0;
    for (int s = t; s < SEQ; s += 256) {
        if (inst[b * SEQ + s]) { if (s < fi) fi = s; ni++; }
        if (pad [b * SEQ + s]) { if (s < fp) fp = s; }
    }
    atomicMin(&s_fi, fi);
    atomicMin(&s_fp, fp);
    atomicAdd(&s_ni, ni);
    __syncthreads();
    if (t == 0) {
        params[b * 2 + 0] = s_fp;           // pad_len
        params[b * 2 + 1] = s_fi + s_ni;    // thresh
    }
}

// ---------------------------------------------------------------------------
// Kernel 2: xbar[b,s,d] = mean of x[b, pad_len .. max(thresh,s+1)-1, d]
// (uniform-softmax token mixing), 0 on pad rows.
// ---------------------------------------------------------------------------
__global__ void k_xbar(const float* __restrict__ x, const int* __restrict__ params,
                       float* __restrict__ xbar) {
    int b = blockIdx.y;
    int d = blockIdx.x * 256 + threadIdx.x;
    int pl = params[b * 2 + 0];
    int th = params[b * 2 + 1];
    const float* xb = x    + (size_t)b * SEQ * DIN + d;
    float*       ob = xbar + (size_t)b * SEQ * DIN + d;
    float acc = 0.f;
    for (int s = pl; s < th; ++s) acc += xb[(size_t)s * DIN];
    float v0 = acc / (float)(th - pl);
    for (int s = 0; s < th; ++s)
        ob[(size_t)s * DIN] = (s >= pl) ? v0 : 0.f;
    for (int s = th; s < SEQ; ++s) {
        acc += xb[(size_t)s * DIN];
        ob[(size_t)s * DIN] = acc / (float)(s + 1 - pl);
    }
}

// ---------------------------------------------------------------------------
// Kernel 3: C[token,k] = softmax_k( x[token,:] . phi[:,k] )
// ---------------------------------------------------------------------------
__global__ void k_router(const float* __restrict__ x, const float* __restrict__ phi,
                         float* __restrict__ Cw) {
    int tok = blockIdx.x;
    int t = threadIdx.x;                       // 128 threads
    __shared__ float red[128][NSK];
    float part[NSK];
#pragma unroll
    for (int k = 0; k < NSK; ++k) part[k] = 0.f;
    const float* xr = x + (size_t)tok * DIN;
    for (int d = t; d < DIN; d += 128) {
        float xv = xr[d];
        const float* pr = phi + (size_t)d * NSK;
#pragma unroll
        for (int k = 0; k < NSK; ++k) part[k] += xv * pr[k];
    }
#pragma unroll
    for (int k = 0; k < NSK; ++k) red[t][k] = part[k];
    __syncthreads();
    for (int off = 64; off > 0; off >>= 1) {
        if (t < off) {
#pragma unroll
            for (int k = 0; k < NSK; ++k) red[t][k] += red[t + off][k];
        }
        __syncthreads();
    }
    if (t == 0) {
        float m = red[0][0];
#pragma unroll
        for (int k = 1; k < NSK; ++k) m = fmaxf(m, red[0][k]);
        float e[NSK], ssum = 0.f;
#pragma unroll
        for (int k = 0; k < NSK; ++k) { e[k] = expf(red[0][k] - m); ssum += e[k]; }
        float inv = 1.f / ssum;
#pragma unroll
        for (int k = 0; k < NSK; ++k) Cw[(size_t)tok * NSK + k] = e[k] * inv;
    }
}

// ---------------------------------------------------------------------------
// Kernel 4: u[token, k*4+r] = C[token,k] * sum_d xbar[token,d]*lora_a[k,d,r]
// ---------------------------------------------------------------------------
__global__ void k_u(const float* __restrict__ xbar, const float* __restrict__ la,
                    const float* __restrict__ Cw, float* __restrict__ u) {
    int tok = blockIdx.x;
    int t = threadIdx.x;                       // 128 threads
    __shared__ float red[128][NKR];            // 16 KB
    float part[NKR];
#pragma unroll
    for (int i = 0; i < NKR; ++i) part[i] = 0.f;
    const float* xr = xbar + (size_t)tok * DIN;
    for (int d = t; d < DIN; d += 128) {
        float xv = xr[d];
#pragma unroll
        for (int k = 0; k < NSK; ++k) {
            const float* ar = la + (size_t)k * DIN * RANK + (size_t)d * RANK;
#pragma unroll
            for (int r = 0; r < RANK; ++r) part[k * RANK + r] += xv * ar[r];
        }
    }
#pragma unroll
    for (int i = 0; i < NKR; ++i) red[t][i] = part[i];
    __syncthreads();
    for (int off = 64; off > 0; off >>= 1) {
        if (t < off) {
#pragma unroll
            for (int i = 0; i < NKR; ++i) red[t][i] += red[t + off][i];
        }
        __syncthreads();
    }
    if (t < NKR)
        u[(size_t)tok * NKR + t] = Cw[(size_t)tok * NSK + (t >> 2)] * red[0][t];
}

// ---------------------------------------------------------------------------
// Tensor Data Mover: DMA one 64x64 f32 tile (row stride 1024 floats) from
// global into LDS, with 4-dword padding every 64 dwords (=> 68-float rows).
// D# per CDNA5 ISA.  Tracked by TENSORcnt.
// ---------------------------------------------------------------------------
__device__ __forceinline__ void tdm_load_tile(unsigned lds_byte, const float* gsrc) {
    unsigned long long ga = (unsigned long long)(uintptr_t)gsrc;
    v4u g0;
    g0[0] = 1u;                                            // count=1, user descriptor
    g0[1] = lds_byte;                                      // lds_addr (bytes)
    g0[2] = (unsigned)(ga & 0xFFFFFFFFu);                  // global_addr[31:0]
    g0[3] = (unsigned)((ga >> 32) & 0x01FFFFFFu)           // global_addr[56:32]
          | (2u << 30);                                    // type = 2 (image)
    v8i g1;
    g1[0] = (2 << 16)                                      // data_size = 4B
          | (1 << 20)                                      // pad_enable
          | (5 << 22)                                      // pad_interval: 64 dwords
          | (3 << 25);                                     // pad_amount: 4 dwords
    g1[1] = (int)((DIN & 0xFFFF) << 16);                   // tensor_dim0[15:0]
    g1[2] = (int)(((unsigned)DIN >> 16)                    // tensor_dim0[31:16]
          | ((NTOK & 0xFFFF) << 16));                      // tensor_dim1[15:0]
    g1[3] = (int)(((unsigned)NTOK >> 16)                   // tensor_dim1[31:16]
          | (64u << 16));                                  // tile_dim0 = 64
    g1[4] = 64;                                            // tile_dim1=64, tile_dim2=0
    g1[5] = DIN;                                           // tensor_dim0_stride[31:0]
    g1[6] = 0;                                             // stride0 hi, dim1_stride lo
    g1[7] = 0;
    v4i g2 = {0, 0, 0, 0};                                 // 2-D tensor: unused
    v4i g3 = {0, 0, 0, 0};
#if __clang_major__ >= 23
    v8i gz = {0, 0, 0, 0, 0, 0, 0, 0};
    __builtin_amdgcn_tensor_load_to_lds(g0, g1, g2, g3, gz, 0);
#else
    __builtin_amdgcn_tensor_load_to_lds(g0, g1, g2, g3, 0);
#endif
}

// ---------------------------------------------------------------------------
// Kernel 5: out[token,o] = x[token,:].W[o,:] + b[o] + u[token,:].lora_b[:,o]
// 64x64 output tiles; TDM double-buffered LDS staging; f32 WMMA main loop.
// ---------------------------------------------------------------------------
__global__ __launch_bounds__(256) void k_gemm(
        const float* __restrict__ x, const float* __restrict__ W,
        const float* __restrict__ bias, const float* __restrict__ u,
        const float* __restrict__ lb, float* __restrict__ out) {
    __shared__ float Xs [2][64][68];   // double-buffered token x k panel
    __shared__ float Wsh[2][64][68];   // double-buffered output x k panel
    __shared__ float Us [64][NKR];     // adapter u tile
    __shared__ float Bsh[NKR][64];     // lora_b tile

    int t = threadIdx.x;
    int mBase = blockIdx.x * 64;       // token tile (flat [4096] tokens)
    int nBase = blockIdx.y * 64;       // output-column tile

    // Epilogue tiles (independent of K loop)
    for (int i = t; i < 64 * NKR; i += 256)
        Us[i / NKR][i % NKR] = u[(size_t)(mBase + i / NKR) * NKR + (i % NKR)];
    for (int i = t; i < NKR * 64; i += 256)
        Bsh[i / 64][i % 64] = lb[(size_t)(i / 64) * DOUT + nBase + (i % 64)];

    int wave = t >> 5, lane = t & 31;
    int half = lane >> 4, lm = lane & 15;
    int mt = wave & 3;                 // M subtile 0..3
    int nh = wave >> 2;                // N half 0..1
    bool issuer = (t < 32);            // wave 0 drives the TDM

    unsigned xs0 = (unsigned)(uintptr_t)&Xs [0][0][0];
    unsigned ws0 = (unsigned)(uintptr_t)&Wsh[0][0][0];
    const unsigned bufBytes = 64u * 68u * 4u;
    const float* xcorner = x + (size_t)mBase * DIN;
    const float* wcorner = W + (size_t)nBase * DIN;

    if (issuer) {                      // prime buffer 0 with K-chunk 0
        tdm_load_tile(xs0, xcorner);
        tdm_load_tile(ws0, wcorner);
    }

    v8f c0 = {};
    v8f c1 = {};
    int buf = 0;

    for (int kb = 0; kb < DIN; kb += 64) {
        if (issuer) __builtin_amdgcn_s_wait_tensorcnt(0);  // chunk kb resident
        __syncthreads();                                   // publish to all waves
        int nkb = kb + 64;
        if (issuer && nkb < DIN) {                         // prefetch next chunk
            tdm_load_tile(xs0 + (buf ^ 1) * bufBytes, xcorner + nkb);
            tdm_load_tile(ws0 + (buf ^ 1) * bufBytes, wcorner + nkb);
        }
#pragma unroll
        for (int kk = 0; kk < 64; kk += 4) {
            int k0 = kk + half * 2;
            // A frag: 16x4, lanes 0-15 K={k0,k0+1}, lanes 16-31 K={k0+2,k0+3}
            v2f a;
            a.x = Xs[buf][mt * 16 + lm][k0];
            a.y = Xs[buf][mt * 16 + lm][k0 + 1];
            v2f b0v, b1v;
            b0v.x = Wsh[buf][nh * 32 + lm][k0];
            b0v.y = Wsh[buf][nh * 32 + lm][k0 + 1];
            b1v.x = Wsh[buf][nh * 32 + 16 + lm][k0];
            b1v.y = Wsh[buf][nh * 32 + 16 + lm][k0 + 1];
            c0 = __builtin_amdgcn_wmma_f32_16x16x4_f32(false, a, false, b0v,
                                                       (short)0, c0, false, false);
            c1 = __builtin_amdgcn_wmma_f32_16x16x4_f32(false, a, false, b1v,
                                                       (short)0, c1, false, false);
        }
        buf ^= 1;
    }

    // Epilogue: acc += u-tile . lora_b-tile, + bias, store.
    int ncol0 = nh * 32 + lm;
    int ncol1 = nh * 32 + 16 + lm;
    float acc0[8], acc1[8];
#pragma unroll
    for (int v = 0; v < 8; ++v) { acc0[v] = c0[v]; acc1[v] = c1[v]; }
#pragma unroll 4
    for (int kr = 0; kr < NKR; ++kr) {
        float bc0 = Bsh[kr][ncol0];
        float bc1 = Bsh[kr][ncol1];
#pragma unroll
        for (int v = 0; v < 8; ++v) {
            float uv = Us[mt * 16 + half * 8 + v][kr];
            acc0[v] += uv * bc0;
            acc1[v] += uv * bc1;
        }
    }
    float bv0 = bias[nBase + ncol0];
    float bv1 = bias[nBase + ncol1];
#pragma unroll
    for (int v = 0; v < 8; ++v) {
        int m = mBase + mt * 16 + half * 8 + v;   // C/D layout: VGPR v -> M = v + 8*half
        out[(size_t)m * DOUT + nBase + ncol0] = acc0[v] + bv0;
        out[(size_t)m * DOUT + nBase + ncol1] = acc1[v] + bv1;
    }
}

// ---------------------------------------------------------------------------
extern "C" void kernel_launch(void* const* d_in, const int* in_sizes, int n_in,
                              void* d_out, int out_size, void* d_ws, size_t ws_size,
                              hipStream_t stream) {
    const float* x    = (const float*)d_in[0];
    const float* phi  = (const float*)d_in[1];
    const float* la   = (const float*)d_in[2];
    const float* lb   = (const float*)d_in[3];
    const float* W    = (const float*)d_in[4];
    const float* bias = (const float*)d_in[5];
    const int*   inst = (const int*)d_in[6];
    const int*   pad  = (const int*)d_in[7];
    float* out = (float*)d_out;

    // workspace layout: [params: 8 ints][xbar: 4096*1024 f32][C: 4096*8][u: 4096*32]
    int*   params = (int*)d_ws;
    float* xbar   = (float*)((char*)d_ws + 256);
    float* Cw     = xbar + (size_t)NTOK * DIN;
    float* u      = Cw   + (size_t)NTOK * NSK;

    k_params<<<BS, 256, 0, stream>>>(inst, pad, params);
    dim3 g2(DIN / 256, BS);
    k_xbar<<<g2, 256, 0, stream>>>(x, params, xbar);
    k_router<<<NTOK, 128, 0, stream>>>(x, phi, Cw);
    k_u<<<NTOK, 128, 0, stream>>>(xbar, la, Cw, u);
    dim3 g5(NTOK / 64, DOUT / 64);
    k_gemm<<<g5, 256, 0, stream>>>(x, W, bias, u, lb, out);
}